// BidirectionalMamba_31361851195874
// MI455X (gfx1250) — compile-verified
//
#include <hip/hip_runtime.h>
#include <math.h>

// ---------------- problem constants ----------------
#define B_   2
#define S_   2048
#define DM_  256
#define DI_  512
#define N_   16
#define R_   32
#define K_   4
#define ROWS_ (B_ * S_)   // 4096 token rows

typedef float v2f __attribute__((ext_vector_type(2)));
typedef float v8f __attribute__((ext_vector_type(8)));
typedef unsigned int u32x4 __attribute__((ext_vector_type(4)));
typedef int i32x4 __attribute__((ext_vector_type(4)));
typedef int i32x8 __attribute__((ext_vector_type(8)));

__device__ __forceinline__ float silu_f(float v) {
  return v * (1.0f / (1.0f + __expf(-v)));
}
__device__ __forceinline__ float softplus_f(float v) {
  return fmaxf(v, 0.0f) + log1pf(__expf(-fabsf(v)));
}

// ---------------- Tensor Data Mover: 2D tile Global -> LDS ----------------
// Builds a D# per CDNA5 ISA:
//  group0: count=1 | lds_addr | global_addr(57b) | type=2
//  group1: data_size=4B, pad fields, tensor dims, tile dims, dim0 stride
// pad=1 inserts 1 DWORD of LDS padding every 64 DWORDs (pad_interval code 5)
// so 64-wide f32 tile rows land at stride 65 -> bank-conflict-free reads.
__device__ __forceinline__ void tdm_load_2d(const float* gsrc, unsigned lds_off,
                                            unsigned td0, unsigned td1,
                                            unsigned tile0, unsigned tile1,
                                            unsigned long long stride0, int pad) {
  unsigned long long ga = (unsigned long long)(size_t)gsrc;
  u32x4 g0 = {0u, 0u, 0u, 0u};
  g0.x = 1u;                                   // count=1 (valid user D#)
  g0.y = lds_off;                              // lds_addr (bytes)
  g0.z = (unsigned)ga;                         // global_addr[31:0]
  g0.w = (unsigned)(ga >> 32) | 0x80000000u;   // global_addr[56:32] | type=2
  int w0 = (2 << 16);                          // data_size = 4 bytes
  if (pad) w0 |= (1 << 20) | (5 << 22);        // pad_enable, every 64 DW, +1 DW
  i32x8 g1;
  g1[0] = w0;
  g1[1] = (int)((td0 & 0xffffu) << 16);                       // tensor_dim0 lo
  g1[2] = (int)((td0 >> 16) | ((td1 & 0xffffu) << 16));       // d0 hi | d1 lo
  g1[3] = (int)((td1 >> 16) | (tile0 << 16));                 // d1 hi | tile0
  g1[4] = (int)tile1;                                         // tile1 | tile2=0
  g1[5] = (int)(unsigned)(stride0 & 0xffffffffull);           // dim0 stride lo
  g1[6] = (int)(unsigned)(stride0 >> 32);                     // stride hi|d1s lo
  g1[7] = 0;
  i32x4 z4 = {0, 0, 0, 0};
#if __has_include(<hip/amd_detail/amd_gfx1250_TDM.h>)
  i32x8 z8 = {0, 0, 0, 0, 0, 0, 0, 0};
  __builtin_amdgcn_tensor_load_to_lds(g0, g1, z4, z4, z8, 0);
#else
  __builtin_amdgcn_tensor_load_to_lds(g0, g1, z4, z4, 0);
#endif
}

// ---------------- LayerNorm over DM=256 (one row per workgroup) ----------------
__global__ __launch_bounds__(DM_) void ln_kernel(const float* __restrict__ x,
                                                 const float* __restrict__ g,
                                                 const float* __restrict__ b,
                                                 float* __restrict__ xn) {
  __shared__ float s1[DM_], s2[DM_];
  const int row = blockIdx.x, t = threadIdx.x;
  const float v = x[(size_t)row * DM_ + t];
  s1[t] = v; s2[t] = v * v;
  __syncthreads();
  for (int off = DM_ / 2; off > 0; off >>= 1) {
    if (t < off) { s1[t] += s1[t + off]; s2[t] += s2[t + off]; }
    __syncthreads();
  }
  const float mean = s1[0] * (1.0f / DM_);
  const float var  = s2[0] * (1.0f / DM_) - mean * mean;
  const float inv  = rsqrtf(var + 1e-5f);
  xn[(size_t)row * DM_ + t] = (v - mean) * inv * g[t] + b[t];
}

// ======= TDM+LDS staged, double-buffered f32 WMMA GEMM (N % 64 == 0) =======
// Block = 4 waves = one 16(row) x 64(col) macro-tile. Software pipeline:
// wave 0 issues the TDM descriptor pair for chunk c+1 into the alternate LDS
// buffer, then `s_wait_tensorcnt 2` (TENSORcnt is in-order per wave, so <=2
// outstanding guarantees chunk c's pair has landed while c+1's DMA overlaps
// the 16 WMMA steps). All 4 waves feed V_WMMA_F32_16X16X4_F32 from LDS.
// EPI: 0 store, 1 +extra[row,col] (residual/accumulate), 2 softplus(+bias col)
template <int EPI>
__global__ __launch_bounds__(128) void wmma_gemm_tdm(const float* __restrict__ A,
                                                     const float* __restrict__ W,
                                                     float* __restrict__ C,
                                                     const float* __restrict__ extra,
                                                     int M, int N, int K) {
  __shared__ float A_s[2][16 * 65];   // ping-pong 16 x (64 + pad) A chunks
  __shared__ float W_s[2][64 * 65];   // ping-pong 64 x (64 + pad) W chunks
  const int lane = threadIdx.x & 31;
  const int wave = threadIdx.x >> 5;
  const int ncb = N >> 6;                      // 64-col macro tiles
  const int row0 = (blockIdx.x / ncb) << 4;
  const int cb0  = (blockIdx.x % ncb) << 6;    // macro-tile col base
  const int col0 = cb0 + (wave << 4);          // this wave's 16 cols
  const int l15 = lane & 15;
  const int kh  = (lane >> 4) << 1;            // K sub-offset within x4 step
  const int nk  = (K + 63) >> 6;               // number of 64-deep K chunks

  auto issue_chunk = [&](int c, int buf) {
    const int k0 = c << 6;
    const int kc = (K - k0 < 64) ? (K - k0) : 64;
    const int apad = (kc == 64) ? 1 : 0;
    tdm_load_2d(A + (size_t)row0 * K + k0, (unsigned)(size_t)(&A_s[buf][0]),
                kc, 16, kc, 16, (unsigned long long)K, apad);
    tdm_load_2d(W + (size_t)k0 * N + cb0, (unsigned)(size_t)(&W_s[buf][0]),
                64, kc, 64, kc, (unsigned long long)N, 1);
  };

  v8f acc = {0.f, 0.f, 0.f, 0.f, 0.f, 0.f, 0.f, 0.f};
  if (wave == 0) issue_chunk(0, 0);            // prologue DMA
  for (int c = 0; c < nk; ++c) {
    const int buf = c & 1;
    const int kc = (K - (c << 6) < 64) ? (K - (c << 6)) : 64;
    const int apad = (kc == 64) ? 1 : 0;
    if (wave == 0) {
      if (c + 1 < nk) {
        issue_chunk(c + 1, buf ^ 1);           // overlap next chunk's DMA
        __builtin_amdgcn_s_wait_tensorcnt(2);  // chunk c landed; c+1 in flight
      } else {
        __builtin_amdgcn_s_wait_tensorcnt(0);  // drain on last chunk
      }
    }
    __syncthreads();
    const int astr = apad ? 65 : kc;           // TDM packs; pad makes 65
    const float* __restrict__ arow = &A_s[buf][0] + l15 * astr;
    const float* __restrict__ wcol = &W_s[buf][0] + (wave << 4) + l15;
#pragma unroll 4
    for (int kk = 0; kk < kc; kk += 4) {
      const int ka = kk + kh;
      v2f a  = {arow[ka], arow[ka + 1]};
      v2f bm = {wcol[ka * 65], wcol[(ka + 1) * 65]};
      acc = __builtin_amdgcn_wmma_f32_16x16x4_f32(false, a, false, bm,
                                                  (short)0, acc, false, false);
    }
    __syncthreads();                           // buf free for chunk c+2 DMA
  }
  const int mrow0 = row0 + ((lane >> 4) << 3);
  const int ccol  = col0 + l15;
#pragma unroll
  for (int r = 0; r < 8; ++r) {
    const size_t idx = (size_t)(mrow0 + r) * N + ccol;
    float v = acc[r];
    if (EPI == 1) v += extra[idx];
    if (EPI == 2) v = softplus_f(v + extra[ccol]);
    C[idx] = v;
  }
}

// ======= global-fed f32 WMMA GEMM (small N: 16/32) =======
template <int EPI>
__global__ __launch_bounds__(128) void wmma_gemm_f32(const float* __restrict__ A,
                                                     const float* __restrict__ W,
                                                     float* __restrict__ C,
                                                     const float* __restrict__ extra,
                                                     int M, int N, int K) {
  const int lane = threadIdx.x & 31;
  const int wave = threadIdx.x >> 5;
  const int ntn = N >> 4;
  const int total = (M >> 4) * ntn;
  const int t = blockIdx.x * 4 + wave;
  if (t >= total) return;
  const int row0 = (t / ntn) << 4;
  const int col0 = (t % ntn) << 4;
  const int l15 = lane & 15;
  const int kh  = (lane >> 4) << 1;
  const float* __restrict__ arow = A + (size_t)(row0 + l15) * K;
  const float* __restrict__ wcol = W + (col0 + l15);
  v8f acc = {0.f, 0.f, 0.f, 0.f, 0.f, 0.f, 0.f, 0.f};
#pragma unroll 4
  for (int k = 0; k < K; k += 4) {
    const int ka = k + kh;
    v2f a  = {arow[ka], arow[ka + 1]};
    v2f bm = {wcol[(size_t)ka * N], wcol[(size_t)(ka + 1) * N]};
    acc = __builtin_amdgcn_wmma_f32_16x16x4_f32(false, a, false, bm,
                                                (short)0, acc, false, false);
  }
  const int mrow0 = row0 + ((lane >> 4) << 3);
  const int ccol  = col0 + l15;
#pragma unroll
  for (int r = 0; r < 8; ++r) {
    const size_t idx = (size_t)(mrow0 + r) * N + ccol;
    float v = acc[r];
    if (EPI == 1) v += extra[idx];
    if (EPI == 2) v = softplus_f(v + extra[ccol]);
    C[idx] = v;
  }
}

// ---------------- causal depthwise conv (K=4) + SiLU ----------------
__global__ __launch_bounds__(DI_) void conv_silu_kernel(const float* __restrict__ xz,
                                                        const float* __restrict__ cw,
                                                        const float* __restrict__ cb,
                                                        float* __restrict__ xc) {
  const int d = threadIdx.x, s = blockIdx.x, b = blockIdx.y;
  float acc = cb[d];
#pragma unroll
  for (int k = 0; k < K_; ++k) {
    const int sp = s - (K_ - 1) + k;
    if (sp >= 0)
      acc += cw[d * K_ + k] * xz[(size_t)(b * S_ + sp) * (2 * DI_) + d];
  }
  xc[(size_t)(b * S_ + s) * DI_ + d] = silu_f(acc);
}

// ---------------- sequential SSM scan --------------------------------
// One (b,d) channel per 16-lane group; lane owns state n. Per step:
// h = exp(dl*A[d,n])*h + (dl*x)*B[n];  y = sum_n h*C[n] via shfl_xor butterfly.
__global__ __launch_bounds__(512) void ssm_scan_kernel(const float* __restrict__ delta,
                                                       const float* __restrict__ xc,
                                                       const float* __restrict__ Bm,
                                                       const float* __restrict__ Cm,
                                                       const float* __restrict__ A_log,
                                                       const float* __restrict__ Dp,
                                                       float* __restrict__ y) {
  const int tid = threadIdx.x;
  const int n = tid & 15;
  const int d = blockIdx.x * 32 + (tid >> 4);
  const int b = blockIdx.y;
  const float Ad = -__expf(A_log[d * N_ + n]);
  const float Dd = Dp[d];
  float h = 0.0f;
  const size_t base = (size_t)b * S_;
  for (int s = 0; s < S_; ++s) {
    const size_t r = base + s;
    const float dl = delta[r * DI_ + d];
    const float xv = xc[r * DI_ + d];
    const float Bv = Bm[r * N_ + n];
    const float Cv = Cm[r * N_ + n];
    h = __expf(dl * Ad) * h + (dl * xv) * Bv;
    float p = h * Cv;
    p += __shfl_xor(p, 1, 32);
    p += __shfl_xor(p, 2, 32);
    p += __shfl_xor(p, 4, 32);
    p += __shfl_xor(p, 8, 32);
    if (n == 0) y[r * DI_ + d] = p + Dd * xv;
    if (s + 1 < S_) {                      // global_prefetch_b8 for next step
      if (n == 0) {
        __builtin_prefetch(&delta[(r + 1) * DI_ + d], 0, 0);
        __builtin_prefetch(&xc[(r + 1) * DI_ + d], 0, 0);
      } else if (n == 1) {
        __builtin_prefetch(&Bm[(r + 1) * N_], 0, 0);
        __builtin_prefetch(&Cm[(r + 1) * N_], 0, 0);
      }
    }
  }
}

// ---------------- gate: yg = y * silu(z) ----------------
__global__ __launch_bounds__(256) void gate_kernel(const float* __restrict__ y,
                                                   const float* __restrict__ xz,
                                                   float* __restrict__ yg) {
  const size_t idx = (size_t)blockIdx.x * 256 + threadIdx.x;  // over ROWS_*DI_
  const size_t row = idx >> 9;
  const int d = (int)(idx & (DI_ - 1));
  const float z = xz[row * (2 * DI_) + DI_ + d];
  yg[idx] = y[idx] * silu_f(z);
}

// ---------------- time reversal of a [B,S,DM] tensor ----------------
__global__ __launch_bounds__(256) void reverse_kernel(const float* __restrict__ src,
                                                      float* __restrict__ dst) {
  const size_t idx = (size_t)blockIdx.x * 256 + threadIdx.x;  // over ROWS_*DM_
  const int c = (int)(idx & (DM_ - 1));
  const size_t row = idx >> 8;
  const int s = (int)(row & (S_ - 1));
  const size_t b = row >> 11;
  dst[idx] = src[((b << 11) + (size_t)(S_ - 1 - s)) * DM_ + c];
}

// ---------------- host orchestration ----------------
extern "C" void kernel_launch(void* const* d_in, const int* in_sizes, int n_in,
                              void* d_out, int out_size, void* d_ws, size_t ws_size,
                              hipStream_t stream) {
  (void)in_sizes; (void)n_in; (void)out_size; (void)ws_size;
  const float* x       = (const float*)d_in[0];
  const float* in_w    = (const float*)d_in[1];
  const float* conv_w  = (const float*)d_in[2];
  const float* conv_b  = (const float*)d_in[3];
  const float* A_log   = (const float*)d_in[4];
  const float* xd_w    = (const float*)d_in[5];
  const float* xB_w    = (const float*)d_in[6];
  const float* xC_w    = (const float*)d_in[7];
  const float* dtp_w   = (const float*)d_in[8];
  const float* dtp_b   = (const float*)d_in[9];
  const float* Dp      = (const float*)d_in[10];
  const float* out_w   = (const float*)d_in[11];
  const float* ln_g    = (const float*)d_in[12];
  const float* ln_b    = (const float*)d_in[13];
  const float* merge_w = (const float*)d_in[14];
  float* out = (float*)d_out;

  char* ws = (char*)d_ws;
  size_t off = 0;
  auto alloc = [&](size_t elems) {
    float* p = (float*)(ws + off);
    off += ((elems * sizeof(float)) + 255) & ~(size_t)255;
    return p;
  };
  float* xrev  = alloc((size_t)ROWS_ * DM_);
  float* hfa   = alloc((size_t)ROWS_ * DM_);
  float* hfb   = alloc((size_t)ROWS_ * DM_);
  float* hba   = alloc((size_t)ROWS_ * DM_);
  float* hbb   = alloc((size_t)ROWS_ * DM_);
  float* hrev  = alloc((size_t)ROWS_ * DM_);
  float* xn    = alloc((size_t)ROWS_ * DM_);
  float* xz    = alloc((size_t)ROWS_ * 2 * DI_);
  float* xc    = alloc((size_t)ROWS_ * DI_);
  float* dtmp  = alloc((size_t)ROWS_ * R_);
  float* delta = alloc((size_t)ROWS_ * DI_);   // reused as yg after the scan
  float* Bmb   = alloc((size_t)ROWS_ * N_);
  float* Cmb   = alloc((size_t)ROWS_ * N_);
  float* ybuf  = alloc((size_t)ROWS_ * DI_);

  auto gblocks  = [](int M, int N) { return ((M >> 4) * (N >> 4) + 3) >> 2; };
  auto tblocks  = [](int M, int N) { return (M >> 4) * (N >> 6); };

  auto run_block = [&](int i, const float* h_in, float* h_out) {
    ln_kernel<<<ROWS_, DM_, 0, stream>>>(h_in, ln_g + i * DM_, ln_b + i * DM_, xn);
    wmma_gemm_tdm<0><<<tblocks(ROWS_, 2 * DI_), 128, 0, stream>>>(
        xn, in_w + (size_t)i * DM_ * 2 * DI_, xz, nullptr, ROWS_, 2 * DI_, DM_);
    conv_silu_kernel<<<dim3(S_, B_), DI_, 0, stream>>>(
        xz, conv_w + (size_t)i * DI_ * K_, conv_b + (size_t)i * DI_, xc);
    wmma_gemm_f32<0><<<gblocks(ROWS_, R_), 128, 0, stream>>>(
        xc, xd_w + (size_t)i * DI_ * R_, dtmp, nullptr, ROWS_, R_, DI_);
    wmma_gemm_tdm<2><<<tblocks(ROWS_, DI_), 128, 0, stream>>>(
        dtmp, dtp_w + (size_t)i * R_ * DI_, delta, dtp_b + (size_t)i * DI_,
        ROWS_, DI_, R_);
    wmma_gemm_f32<0><<<gblocks(ROWS_, N_), 128, 0, stream>>>(
        xc, xB_w + (size_t)i * DI_ * N_, Bmb, nullptr, ROWS_, N_, DI_);
    wmma_gemm_f32<0><<<gblocks(ROWS_, N_), 128, 0, stream>>>(
        xc, xC_w + (size_t)i * DI_ * N_, Cmb, nullptr, ROWS_, N_, DI_);
    ssm_scan_kernel<<<dim3(DI_ / 32, B_), 512, 0, stream>>>(
        delta, xc, Bmb, Cmb, A_log + (size_t)i * DI_ * N_, Dp + (size_t)i * DI_,
        ybuf);
    gate_kernel<<<(ROWS_ * DI_) / 256, 256, 0, stream>>>(ybuf, xz, delta);
    wmma_gemm_tdm<1><<<tblocks(ROWS_, DM_), 128, 0, stream>>>(
        delta, out_w + (size_t)i * DI_ * DM_, h_out, h_in, ROWS_, DM_, DI_);
  };

  // forward chain: blocks 0,1
  run_block(0, x, hfa);
  run_block(1, hfa, hfb);
  // backward chain: reverse time, blocks 2,3, reverse back
  reverse_kernel<<<(ROWS_ * DM_) / 256, 256, 0, stream>>>(x, xrev);
  run_block(2, xrev, hba);
  run_block(3, hba, hbb);
  reverse_kernel<<<(ROWS_ * DM_) / 256, 256, 0, stream>>>(hbb, hrev);
  // merge: out = fwd @ merge_w[:DM] + bwd @ merge_w[DM:]
  wmma_gemm_tdm<0><<<tblocks(ROWS_, DM_), 128, 0, stream>>>(
      hfb, merge_w, out, nullptr, ROWS_, DM_, DM_);
  wmma_gemm_tdm<1><<<tblocks(ROWS_, DM_), 128, 0, stream>>>(
      hrev, merge_w + (size_t)DM_ * DM_, out, out, ROWS_, DM_, DM_);
}